// NoCarryOverSRNetLayer_80805514707500
// MI455X (gfx1250) — compile-verified
//
#include <hip/hip_runtime.h>
#include <hip/hip_bf16.h>

// ---------------------------------------------------------------------------
// Problem constants (from the reference)
// ---------------------------------------------------------------------------
#define NUM_LAYERS 2
#define H 8
#define D 64
#define HID (H * D)      // 512
#define FF 2048
#define SLEN 256
#define BSZ 16
#define IN_DIM 512
#define M_ROWS (SLEN * BSZ)   // 4096

typedef float v2f __attribute__((ext_vector_type(2)));
typedef float v8f __attribute__((ext_vector_type(8)));

// Only the device pass knows AMDGCN builtins; host pass must not see #error.
#if defined(__HIP_DEVICE_COMPILE__) && !__has_builtin(__builtin_amdgcn_wmma_f32_16x16x4_f32)
#error "gfx1250 f32 WMMA builtin __builtin_amdgcn_wmma_f32_16x16x4_f32 not available"
#endif

// ---------------------------------------------------------------------------
// Fused GEMM:  C[M,N] = epilogue( A[M,K] @ B[K,N] )
// epilogue: optional +bias[N], optional ReLU, optional +resid[M,N]
// f32 end-to-end via V_WMMA_F32_16X16X4_F32.
//
// Block = 256 threads (8 waves), block tile 64x64. Waves in a 4x2 grid; each
// wave owns a 16x32 slab = two 16x16 accumulators (A-frag reused across both).
// K staged through LDS in chunks of 32 with float4 (b128) copies; all frags
// for a chunk are pulled into registers first, then 16 WMMAs issue back-to-
// back so LDS waits amortize.
// ---------------------------------------------------------------------------
#define TK 32
#define LDA_S 36   // padded LDS row stride for A tile (floats); 16B-aligned rows
#define LDB_S 68   // padded LDS row stride for B tile (floats); 16B-aligned rows

__global__ void __launch_bounds__(256)
gemm_wmma_f32(const float* __restrict__ A, const float* __restrict__ B,
              const float* __restrict__ bias, const float* __restrict__ resid,
              float* __restrict__ C, int M, int N, int K, int relu)
{
    __shared__ float As[64 * LDA_S];   // 64 x TK
    __shared__ float Bs[TK * LDB_S];   // TK x 64

    const int tid   = threadIdx.x;
    const int lane  = tid & 31;
    const int wave  = tid >> 5;        // 0..7
    const int wr    = wave >> 1;       // 0..3 -> rows 16*wr
    const int wc    = wave & 1;        // 0..1 -> cols 32*wc
    const int mBase = blockIdx.y * 64;
    const int nBase = blockIdx.x * 64;
    const int lm    = lane & 15;
    const int hi    = lane >> 4;       // 0: K=k4+0,1   1: K=k4+2,3

    v8f acc0 = {0.f, 0.f, 0.f, 0.f, 0.f, 0.f, 0.f, 0.f};
    v8f acc1 = {0.f, 0.f, 0.f, 0.f, 0.f, 0.f, 0.f, 0.f};

    for (int kc = 0; kc < K; kc += TK) {
        // --- stage 64xTK A tile and TKx64 B tile via float4 (b128) copies ---
#pragma unroll
        for (int i = 0; i < 2; ++i) {
            const int f4 = tid + i * 256;                 // 0..511
            const int arow = f4 >> 3, acol = (f4 & 7) << 2;
            const float4 av = *(const float4*)&A[(size_t)(mBase + arow) * K + kc + acol];
            *(float4*)&As[arow * LDA_S + acol] = av;
            const int brow = f4 >> 4, bcol = (f4 & 15) << 2;
            const float4 bv = *(const float4*)&B[(size_t)(kc + brow) * N + nBase + bcol];
            *(float4*)&Bs[brow * LDB_S + bcol] = bv;
        }
        // hint next K-tile toward the caches (global_prefetch_b8)
        if (kc + TK < K) {
            __builtin_prefetch(&A[(size_t)(mBase + (tid >> 2)) * K + kc + TK + ((tid & 3) << 3)], 0, 1);
            __builtin_prefetch(&B[(size_t)(kc + TK + (tid >> 4)) * N + nBase + ((tid & 15) << 2)], 0, 1);
        }
        __syncthreads();

#if defined(__HIP_DEVICE_COMPILE__)
        // --- pull all fragments for this K-chunk into registers ---
        v2f af[8], bf0[8], bf1[8];
#pragma unroll
        for (int kk = 0; kk < 8; ++kk) {
            const int ka = kk * 4 + 2 * hi;
            const int arow = (wr * 16 + lm) * LDA_S;
            af[kk].x  = As[arow + ka];
            af[kk].y  = As[arow + ka + 1];
            const int bc = wc * 32 + lm;
            bf0[kk].x = Bs[ka * LDB_S + bc];
            bf0[kk].y = Bs[(ka + 1) * LDB_S + bc];
            bf1[kk].x = Bs[ka * LDB_S + bc + 16];
            bf1[kk].y = Bs[(ka + 1) * LDB_S + bc + 16];
        }
        // --- 16 WMMAs back-to-back (acc0/acc1 interleaved) ---
#pragma unroll
        for (int kk = 0; kk < 8; ++kk) {
            acc0 = __builtin_amdgcn_wmma_f32_16x16x4_f32(
                false, af[kk], false, bf0[kk], (short)0, acc0, false, false);
            acc1 = __builtin_amdgcn_wmma_f32_16x16x4_f32(
                false, af[kk], false, bf1[kk], (short)0, acc1, false, false);
        }
#endif
        __syncthreads();
    }

    // C/D layout: VGPR r, lanes 0-15 -> row +r, lanes 16-31 -> row +8+r
    const int rowBase = mBase + wr * 16 + hi * 8;
    const int col0    = nBase + wc * 32 + lm;
    const int col1    = col0 + 16;
    const float bv0   = bias ? bias[col0] : 0.f;
    const float bv1   = bias ? bias[col1] : 0.f;
#pragma unroll
    for (int r = 0; r < 8; ++r) {
        const int row = rowBase + r;
        float v0 = acc0[r] + bv0;
        float v1 = acc1[r] + bv1;
        if (relu) { v0 = fmaxf(v0, 0.f); v1 = fmaxf(v1, 0.f); }
        if (resid) {
            v0 += resid[(size_t)row * N + col0];
            v1 += resid[(size_t)row * N + col1];
        }
        C[(size_t)row * N + col0] = v0;
        C[(size_t)row * N + col1] = v1;
    }
}

// ---------------------------------------------------------------------------
// SR recurrent scan: one workgroup per (batch, head) pair; fast-weight state
// (Wy,Wq,Wk: 64x64 f32 each, wb: 64x4) lives in LDS for all 256 steps.
// 256 threads: 192 do row dot-products, 4 do beta; rank-1 updates 48/thread.
// (N=1 matvecs: WMMA would run at 1/16 utilization, so VALU is correct here.)
// ---------------------------------------------------------------------------
__device__ __forceinline__ void softmax64(float* v, int tid)
{
    float e = 0.f;
    if (tid < D) {
        float m = -1e30f;
        for (int j = 0; j < D; ++j) m = fmaxf(m, v[j]);   // LDS broadcast reads
        e = __expf(v[tid] - m);
    }
    __syncthreads();
    if (tid < D) v[tid] = e;
    __syncthreads();
    float r = 0.f;
    if (tid < D) {
        float s = 0.f;
        for (int j = 0; j < D; ++j) s += v[j];
        r = v[tid] / s;
    }
    __syncthreads();
    if (tid < D) v[tid] = r;
    __syncthreads();
}

__global__ void __launch_bounds__(256)
sr_scan_kernel(const float* __restrict__ hbuf,   // (SLEN, BSZ, HID)
               const float* __restrict__ Wy0,    // (H, D, D) layer slice
               const float* __restrict__ Wq0,
               const float* __restrict__ Wk0,
               const float* __restrict__ wb0,    // (H, D, 4) layer slice
               float* __restrict__ ybuf)         // (SLEN, BSZ, HID)
{
    __shared__ float sWy[D * D], sWq[D * D], sWk[D * D];
    __shared__ float swb[D * 4];
    __shared__ float sx[D], sy[D], sq[D], sk[D], sd[D];
    __shared__ float sdy[D], sdq[D], sdk[D];
    __shared__ float sbeta[4], sdb[4];

    const int tid = threadIdx.x;
    const int bh  = blockIdx.x;        // b * H + h
    const int b   = bh / H;
    const int hd  = bh % H;

    const float* gWy = Wy0 + (size_t)hd * D * D;
    const float* gWq = Wq0 + (size_t)hd * D * D;
    const float* gWk = Wk0 + (size_t)hd * D * D;
    const float* gwb = wb0 + (size_t)hd * D * 4;

    for (int i = tid; i < D * D; i += 256) {
        sWy[i] = gWy[i];
        sWq[i] = gWq[i];
        sWk[i] = gWk[i];
    }
    for (int i = tid; i < D * 4; i += 256) swb[i] = gwb[i];
    __syncthreads();

    for (int t = 0; t < SLEN; ++t) {
        const float* xp = hbuf + ((size_t)t * BSZ + b) * HID + hd * D;
        if (tid < D) sx[tid] = xp[tid];
        __syncthreads();

        softmax64(sx, tid);   // x = softmax(h_head)

        // Phase 1: y = Wy x ; q' = Wq x ; k' = Wk x ; beta' = wb^T x
        if (tid < 3 * D) {
            const int mat = tid >> 6, row = tid & (D - 1);
            const float* Wr = ((mat == 0) ? sWy : (mat == 1) ? sWq : sWk) + row * D;
            float acc = 0.f;
            for (int j = 0; j < D; ++j) acc = fmaf(Wr[j], sx[j], acc);
            ((mat == 0) ? sy : (mat == 1) ? sq : sk)[row] = acc;
        } else if (tid < 3 * D + 4) {
            const int f = tid - 3 * D;
            float acc = 0.f;
            for (int j = 0; j < D; ++j) acc = fmaf(swb[j * 4 + f], sx[j], acc);
            sbeta[f] = acc;
        }
        __syncthreads();

        // softmax(q) and softmax(k) in parallel thread groups; sigmoid(beta)
        {
            const int grp = tid >> 6;
            const int off = tid & (D - 1);
            float* v = (grp == 0) ? sq : sk;
            float e = 0.f;
            if (grp < 2) {
                float m = -1e30f;
                for (int j = 0; j < D; ++j) m = fmaxf(m, v[j]);
                e = __expf(v[off] - m);
            } else if (tid >= 192 && tid < 196) {
                const int f = tid - 192;
                sbeta[f] = 1.f / (1.f + __expf(-sbeta[f]));
            }
            __syncthreads();
            if (grp < 2) v[off] = e;
            __syncthreads();
            float r = 0.f;
            if (grp < 2) {
                float s = 0.f;
                for (int j = 0; j < D; ++j) s += v[j];
                r = v[off] / s;
            }
            __syncthreads();
            if (grp < 2) v[off] = r;
            __syncthreads();
        }

        if (tid < D) sd[tid] = sq[tid] - sk[tid];
        __syncthreads();

        // Phase 2: dy = Wy d ; dq = Wq d ; dk = Wk d ; db = wb^T d
        if (tid < 3 * D) {
            const int mat = tid >> 6, row = tid & (D - 1);
            const float* Wr = ((mat == 0) ? sWy : (mat == 1) ? sWq : sWk) + row * D;
            float acc = 0.f;
            for (int j = 0; j < D; ++j) acc = fmaf(Wr[j], sd[j], acc);
            ((mat == 0) ? sdy : (mat == 1) ? sdq : sdk)[row] = acc;
        } else if (tid < 3 * D + 4) {
            const int f = tid - 3 * D;
            float acc = 0.f;
            for (int j = 0; j < D; ++j) acc = fmaf(swb[j * 4 + f], sd[j], acc);
            sdb[f] = acc;
        }
        __syncthreads();

        // Emit y (uses pre-update Wy, already computed) + rank-1 updates
        if (tid < D) ybuf[((size_t)t * BSZ + b) * HID + hd * D + tid] = sy[tid];
        const float b0 = sbeta[0], b1 = sbeta[1], b2 = sbeta[2], b3 = sbeta[3];
        for (int i = tid; i < D * D; i += 256) {
            const int r = i >> 6, c = i & (D - 1);
            const float kc = sk[c];
            sWy[i] = fmaf(b0 * sdy[r], kc, sWy[i]);
            sWq[i] = fmaf(b1 * sdq[r], kc, sWq[i]);
            sWk[i] = fmaf(b2 * sdk[r], kc, sWk[i]);
        }
        for (int i = tid; i < D * 4; i += 256) {
            const int j = i >> 2, f = i & 3;
            swb[i] = fmaf(b3 * sk[j], sdb[f], swb[i]);
        }
        __syncthreads();
    }
}

// ---------------------------------------------------------------------------
// LayerNorm over last dim (512), one block per row.
// ---------------------------------------------------------------------------
__global__ void __launch_bounds__(256)
layernorm_kernel(const float* __restrict__ x, const float* __restrict__ g,
                 const float* __restrict__ bb, float* __restrict__ y)
{
    __shared__ float red[256];
    const int row = blockIdx.x;
    const int tid = threadIdx.x;
    const float* xr = x + (size_t)row * HID;

    const float v0 = xr[tid], v1 = xr[tid + 256];
    red[tid] = v0 + v1;
    __syncthreads();
    for (int o = 128; o > 0; o >>= 1) {
        if (tid < o) red[tid] += red[tid + o];
        __syncthreads();
    }
    const float mean = red[0] * (1.f / HID);
    __syncthreads();

    const float c0 = v0 - mean, c1 = v1 - mean;
    red[tid] = c0 * c0 + c1 * c1;
    __syncthreads();
    for (int o = 128; o > 0; o >>= 1) {
        if (tid < o) red[tid] += red[tid + o];
        __syncthreads();
    }
    const float inv = rsqrtf(red[0] * (1.f / HID) + 1e-5f);

    y[(size_t)row * HID + tid]       = c0 * inv * g[tid] + bb[tid];
    y[(size_t)row * HID + tid + 256] = c1 * inv * g[tid + 256] + bb[tid + 256];
}

// ---------------------------------------------------------------------------
// Host-side orchestration
// ---------------------------------------------------------------------------
extern "C" void kernel_launch(void* const* d_in, const int* in_sizes, int n_in,
                              void* d_out, int out_size, void* d_ws, size_t ws_size,
                              hipStream_t stream)
{
    (void)in_sizes; (void)n_in; (void)out_size; (void)ws_size;

    const float* x     = (const float*)d_in[0];   // (SLEN,BSZ,IN_DIM)
    const float* in_w  = (const float*)d_in[1];   // (IN_DIM,HID)
    const float* in_b  = (const float*)d_in[2];   // (HID,)
    const float* Wy    = (const float*)d_in[3];   // (L,H,D,D)
    const float* Wq    = (const float*)d_in[4];
    const float* Wk    = (const float*)d_in[5];
    const float* wb    = (const float*)d_in[6];   // (L,H,D,4)
    const float* Wout  = (const float*)d_in[7];   // (L,HID,HID)
    const float* ln_g  = (const float*)d_in[8];   // (L,HID)
    const float* ln_b  = (const float*)d_in[9];
    const float* ff_w1 = (const float*)d_in[10];  // (L,HID,FF)
    const float* ff_b1 = (const float*)d_in[11];  // (L,FF)
    const float* ff_w2 = (const float*)d_in[12];  // (L,FF,HID)
    const float* ff_b2 = (const float*)d_in[13];  // (L,HID)

    float* out = (float*)d_out;                          // running activation (M_ROWS,HID)
    float* ys  = (float*)d_ws;                           // (M_ROWS,HID): ys, later reused for LN output
    float* ffb = ys + (size_t)M_ROWS * HID;              // (M_ROWS,FF)

    const dim3 gblk(256);
    const dim3 grid_hid(HID / 64, M_ROWS / 64);
    const dim3 grid_ff(FF / 64, M_ROWS / 64);

    // Input projection: out = x @ in_w + in_b
    gemm_wmma_f32<<<grid_hid, gblk, 0, stream>>>(x, in_w, in_b, nullptr, out,
                                                 M_ROWS, HID, IN_DIM, 0);

    for (int l = 0; l < NUM_LAYERS; ++l) {
        // SR block recurrent scan -> ys
        sr_scan_kernel<<<dim3(BSZ * H), dim3(256), 0, stream>>>(
            out,
            Wy + (size_t)l * H * D * D,
            Wq + (size_t)l * H * D * D,
            Wk + (size_t)l * H * D * D,
            wb + (size_t)l * H * D * 4,
            ys);

        // out = out + ys @ Wout[l]
        gemm_wmma_f32<<<grid_hid, gblk, 0, stream>>>(
            ys, Wout + (size_t)l * HID * HID, nullptr, out, out,
            M_ROWS, HID, HID, 0);

        // yln = LN(out)   (reuses ys buffer; ys is dead after Wout GEMM)
        layernorm_kernel<<<dim3(M_ROWS), dim3(256), 0, stream>>>(
            out, ln_g + (size_t)l * HID, ln_b + (size_t)l * HID, ys);

        // ffb = relu(yln @ ff_w1[l] + ff_b1[l])
        gemm_wmma_f32<<<grid_ff, gblk, 0, stream>>>(
            ys, ff_w1 + (size_t)l * HID * FF, ff_b1 + (size_t)l * FF, nullptr, ffb,
            M_ROWS, FF, HID, 1);

        // out = out + ffb @ ff_w2[l] + ff_b2[l]
        gemm_wmma_f32<<<grid_hid, gblk, 0, stream>>>(
            ffb, ff_w2 + (size_t)l * FF * HID, ff_b2 + (size_t)l * HID, out, out,
            M_ROWS, HID, FF, 0);
    }
}